// ObjectSelect_22436909154603
// MI455X (gfx1250) — compile-verified
//
#include <hip/hip_runtime.h>
#include <hip/hip_bf16.h>

// ---------------------------------------------------------------------------
// ObjectSelect for MI455X (gfx1250), compile-only tuning.
//
// Reference reduces to a per-image argmax over 162,000 masked scores (x2
// tensors), then decoding ONE box per image. Memory floor: stream cls_prob
// A+B = 20.7 MB -> ~0.9us @ 23.3 TB/s. bbox_pred (83 MB) is only gathered at
// 32 winner locations. Streaming phase uses the gfx1250 async global->LDS
// path (ASYNCcnt) with an 8-deep pipeline: 7 outstanding b128 copies per
// wave x 2560 waves ~= 9 MB in flight, sized against the HBM
// bandwidth-latency product (23.3 TB/s x ~0.5-1us latency).
// ---------------------------------------------------------------------------

#define BATCH   16
#define NROIS   2000
#define NCLS    81
#define NC      (NROIS * NCLS)       // 162000 floats per image
#define NC4     (NC / 4)             // 40500 float4 per image (16B aligned)
#define THRESH  0.05f
#define NEG_INF_F (-1000000000.0f)

#define BLOCK   256                  // 8 waves (wave32)
#define ITER    16                   // float4 per thread
#define DEPTH   8                    // async pipeline depth (LDS buffers)
#define CHUNK4  (BLOCK * ITER)       // 4096 float4 per block
#define SPLIT   ((NC4 + CHUNK4 - 1) / CHUNK4)   // 10 blocks per image

// The async-to-LDS builtin takes GCC-vector int4 pointers:
//   param0: int4 addrspace(1)*  (global source)
//   param1: int4 addrspace(3)*  (LDS destination)
typedef int v4i __attribute__((vector_size(16)));

// ---- gfx1250 async global->LDS b128 copy (ASYNCcnt tracked) ---------------
__device__ __forceinline__ void async_load_b128(const float4* gsrc, float4* ldst) {
#if __has_builtin(__builtin_amdgcn_global_load_async_to_lds_b128)
    __builtin_amdgcn_global_load_async_to_lds_b128(
        (__attribute__((address_space(1))) v4i*)gsrc,
        (__attribute__((address_space(3))) v4i*)ldst,
        /*offset=*/0, /*cpol=*/0);
#else
    unsigned lds_off = (unsigned)(unsigned long long)ldst;   // low 32b = LDS byte addr
    asm volatile("global_load_async_to_lds_b128 %0, %1, off"
                 :: "v"(lds_off), "v"((unsigned long long)gsrc)
                 : "memory");
#endif
}

template <int N>
__device__ __forceinline__ void wait_asynccnt() {
#if __has_builtin(__builtin_amdgcn_s_wait_asynccnt)
    __builtin_amdgcn_s_wait_asynccnt(N);
#else
    asm volatile("s_wait_asynccnt %0" :: "n"(N) : "memory");
#endif
}

// 64-bit xor-shuffle max across the 32-lane wave
__device__ __forceinline__ unsigned long long shflxor64(unsigned long long v, int m) {
    unsigned lo = (unsigned)v, hi = (unsigned)(v >> 32);
    lo = __shfl_xor(lo, m, 32);
    hi = __shfl_xor(hi, m, 32);
    return ((unsigned long long)hi << 32) | lo;
}

// ---------------------------------------------------------------------------
// Streaming argmax: grid.x = BATCH*SPLIT, grid.y = 2 (A/B).
// (First in file so the disasm snippet shows the async prologue.)
__global__ __launch_bounds__(BLOCK)
void objsel_argmax(const float* __restrict__ clsA,
                   const float* __restrict__ clsB,
                   unsigned long long* __restrict__ ws) {
    __shared__ float4 smem[DEPTH][BLOCK];      // 32 KB, 8-deep async pipeline
    const int tid  = threadIdx.x;
    const int img  = blockIdx.x / SPLIT;
    const int sp   = blockIdx.x % SPLIT;
    const int ten  = blockIdx.y;               // 0 = A, 1 = B

    const float4* src4 =
        (const float4*)((ten == 0 ? clsA : clsB) + (size_t)img * NC);
    const int base4 = sp * CHUNK4 + tid;       // coalesced: stride BLOCK

    unsigned long long best = 0ULL;

    // prologue: fill pipeline with tiles 0..DEPTH-2 (7 outstanding copies)
#pragma unroll
    for (int p = 0; p < DEPTH - 1; ++p) {
        int i4 = base4 + p * BLOCK;
        int g  = i4 < NC4 ? i4 : (NC4 - 1);    // clamp tail (masked later)
        async_load_b128(src4 + g, &smem[p][tid]);
    }

    for (int i = 0; i < ITER; ++i) {
        if (i + DEPTH - 1 < ITER) {
            int i4 = base4 + (i + DEPTH - 1) * BLOCK;
            int g  = i4 < NC4 ? i4 : (NC4 - 1);
            async_load_b128(src4 + g, &smem[(i + DEPTH - 1) & (DEPTH - 1)][tid]);
            wait_asynccnt<DEPTH - 1>();        // in-order: tile i has landed
        } else {
            wait_asynccnt<0>();                // drain at the tail
        }
        const float4 v   = smem[i & (DEPTH - 1)][tid]; // private slot -> no barrier
        const int i4     = base4 + i * BLOCK;
        const bool valid = (i4 < NC4);
        const int flat0  = i4 * 4;
        const float vv[4] = { v.x, v.y, v.z, v.w };
#pragma unroll
        for (int j = 0; j < 4; ++j) {
            const int flat   = flat0 + j;
            const unsigned c = (unsigned)flat % NCLS;   // magic-mul, no div unit
            const float s    = vv[j];
            const bool keep  = valid && (c != 0u) && (s > THRESH);
            const float m    = keep ? s : NEG_INF_F;
            // monotonic float->u32 mapping, then pack with inverted index so
            // u64 max == (max score, then LOWEST flat index) -- matches top_k.
            unsigned su = __float_as_uint(m);
            unsigned mu = (su & 0x80000000u) ? ~su : (su | 0x80000000u);
            unsigned long long key =
                ((unsigned long long)mu << 32) | (0xFFFFFFFFu - (unsigned)flat);
            if (key > best) best = key;
        }
    }

    // wave32 reduction, then one deterministic u64 atomicMax per wave
    for (int m = 16; m > 0; m >>= 1) {
        unsigned long long o = shflxor64(best, m);
        if (o > best) best = o;
    }
    if ((tid & 31) == 0)
        atomicMax(&ws[ten * BATCH + img], best);
}

// ---------------------------------------------------------------------------
__global__ void objsel_init_ws(unsigned long long* __restrict__ ws) {
    int t = threadIdx.x;
    if (t < 2 * BATCH) ws[t] = 0ULL;   // identity for max (all real keys > 0)
}

// One thread per (tensor, image): gather winner, decode + clip box, emit.
__global__ void objsel_finalize(const float* __restrict__ roisA,
                                const float* __restrict__ bbA,
                                const float* __restrict__ infoA,
                                const float* __restrict__ roisB,
                                const float* __restrict__ bbB,
                                const float* __restrict__ infoB,
                                const unsigned long long* __restrict__ ws,
                                float* __restrict__ out) {
    int t = threadIdx.x;
    if (t >= 2 * BATCH) return;
    const int ten = t / BATCH;
    const int b   = t % BATCH;
    const float* rois = ten ? roisB : roisA;
    const float* bb   = ten ? bbB   : bbA;
    const float* info = ten ? infoB : infoA;

    const unsigned long long key = ws[ten * BATCH + b];
    const unsigned mu = (unsigned)(key >> 32);
    const unsigned su = (mu & 0x80000000u) ? (mu & 0x7FFFFFFFu) : ~mu;
    const float score = __uint_as_float(su);
    const unsigned flat = 0xFFFFFFFFu - (unsigned)(key & 0xFFFFFFFFu);
    const unsigned n = flat / NCLS;
    const unsigned c = flat % NCLS;

    const float* r = rois + ((size_t)b * NROIS + n) * 4;
    const float x1 = r[0], y1 = r[1], x2 = r[2], y2 = r[3];
    const float w  = x2 - x1 + 1.0f;
    const float h  = y2 - y1 + 1.0f;
    const float cx = x1 + 0.5f * w;
    const float cy = y1 + 0.5f * h;

    const float* d = bb + ((size_t)b * NROIS + n) * (4 * NCLS) + (size_t)c * 4;
    const float dx = d[0], dy = d[1];
    const float dw = fminf(fmaxf(d[2], -10.0f), 4.0f);
    const float dh = fminf(fmaxf(d[3], -10.0f), 4.0f);

    const float pcx = dx * w + cx;
    const float pcy = dy * h + cy;
    const float pw  = expf(dw) * w;
    const float ph  = expf(dh) * h;

    float ox1 = pcx - 0.5f * pw;
    float oy1 = pcy - 0.5f * ph;
    float ox2 = pcx + 0.5f * pw - 1.0f;
    float oy2 = pcy + 0.5f * ph - 1.0f;

    const float H = info[b * 3 + 0];
    const float W = info[b * 3 + 1];
    ox1 = fminf(fmaxf(ox1, 0.0f), W - 1.0f);
    ox2 = fminf(fmaxf(ox2, 0.0f), W - 1.0f);
    oy1 = fminf(fmaxf(oy1, 0.0f), H - 1.0f);
    oy2 = fminf(fmaxf(oy2, 0.0f), H - 1.0f);

    float* o = out + (size_t)ten * (BATCH * 5) + (size_t)b * 5;
    o[0] = ox1; o[1] = oy1; o[2] = ox2; o[3] = oy2;
    o[4] = fmaxf(score, 0.0f);
    if (ten == 0) out[2 * BATCH * 5 + b] = (float)c;   // sel_id (A winner class)
}

// ---------------------------------------------------------------------------
extern "C" void kernel_launch(void* const* d_in, const int* in_sizes, int n_in,
                              void* d_out, int out_size, void* d_ws, size_t ws_size,
                              hipStream_t stream) {
    const float* roisA = (const float*)d_in[0];
    const float* clsA  = (const float*)d_in[1];
    const float* bbA   = (const float*)d_in[2];
    const float* infoA = (const float*)d_in[3];
    const float* roisB = (const float*)d_in[4];
    const float* clsB  = (const float*)d_in[5];
    const float* bbB   = (const float*)d_in[6];
    const float* infoB = (const float*)d_in[7];
    unsigned long long* ws = (unsigned long long*)d_ws;
    float* out = (float*)d_out;

    hipLaunchKernelGGL(objsel_init_ws, dim3(1), dim3(64), 0, stream, ws);
    hipLaunchKernelGGL(objsel_argmax, dim3(BATCH * SPLIT, 2), dim3(BLOCK), 0, stream,
                       clsA, clsB, ws);
    hipLaunchKernelGGL(objsel_finalize, dim3(1), dim3(64), 0, stream,
                       roisA, bbA, infoA, roisB, bbB, infoB, ws, out);
}